// Classifier_proj_67345087201479
// MI455X (gfx1250) — compile-verified
//
#include <hip/hip_runtime.h>
#include <math.h>

#define N_NODES 100000
#define N_EDGES 1600000
#define DIM 256
#define LDS_PITCH 260   // 16-byte aligned rows, rows step 4 LDS banks -> conflict-free b64 reads

typedef __attribute__((ext_vector_type(2))) float v2f;
typedef __attribute__((ext_vector_type(8))) float v8f;

// ---------------- zero init (d_out used as agg accumulator; ws holds degrees) -------------
__global__ void zero_kernel(float* __restrict__ out, int n_out,
                            float* __restrict__ deg, int n_deg) {
    int i = blockIdx.x * blockDim.x + threadIdx.x;
    if (i < n_out) out[i] = 0.0f;
    if (i < n_deg) deg[i] = 0.0f;
}

// ---------------- degree computation: out_deg[src]++, in_deg[dst]++ -----------------------
__global__ void degree_kernel(const int* __restrict__ src, const int* __restrict__ dst,
                              float* __restrict__ outdeg, float* __restrict__ indeg, int E) {
    int e = blockIdx.x * blockDim.x + threadIdx.x;
    if (e >= E) return;
    unsafeAtomicAdd(&outdeg[src[e]], 1.0f);
    unsafeAtomicAdd(&indeg[dst[e]], 1.0f);
}

// ---------------- deg -> rsqrt(max(deg,1)) in place ---------------------------------------
__global__ void norm_kernel(float* __restrict__ deg, int n) {
    int i = blockIdx.x * blockDim.x + threadIdx.x;
    if (i < n) deg[i] = rsqrtf(fmaxf(deg[i], 1.0f));
}

// ---------------- W[k][c] -> Wt[c][k] so B fragments are contiguous float2 ----------------
__global__ void transpose_kernel(const float* __restrict__ W, float* __restrict__ Wt) {
    int i = blockIdx.x * blockDim.x + threadIdx.x;   // 65536
    int k = i >> 8, c = i & 255;
    Wt[c * DIM + k] = W[k * DIM + c];
}

// ---------------- edge scatter: agg[dst] += h[src] * norm_src[src] ------------------------
// one thread per (edge, 4-float chunk); 64 threads cover one edge's 256 features
__global__ void scatter_kernel(const float4* __restrict__ h4,
                               const int* __restrict__ src, const int* __restrict__ dst,
                               const float* __restrict__ norm_src,
                               float* __restrict__ agg, int E) {
    long long gid = (long long)blockIdx.x * blockDim.x + threadIdx.x;
    int e = (int)(gid >> 6);
    if (e >= E) return;
    int c4 = (int)(gid & 63);
    int s = src[e], d = dst[e];
    float ns = norm_src[s];
    float4 v = h4[(long long)s * 64 + c4];
    float* base = agg + (long long)d * DIM + c4 * 4;
    unsafeAtomicAdd(base + 0, v.x * ns);
    unsafeAtomicAdd(base + 1, v.y * ns);
    unsafeAtomicAdd(base + 2, v.z * ns);
    unsafeAtomicAdd(base + 3, v.w * ns);
}

// ---------------- in-place: out = ELU( (agg * norm_dst) @ W + bias ) ----------------------
// 256 threads = 8 waves; each wave -> 2 column tiles of 16; K=256 in steps of 4 (f32 WMMA)
__global__ __launch_bounds__(256) void gemm_elu_kernel(float* __restrict__ out,
                                                       const float* __restrict__ Wt,
                                                       const float* __restrict__ bias,
                                                       const float* __restrict__ norm_dst) {
    __shared__ float tile[16 * LDS_PITCH];
    const int tid  = threadIdx.x;
    const int row0 = blockIdx.x * 16;

    // stage 16x256 agg tile (scaled by norm_dst) into LDS: 1024 float4, 4 per thread
    #pragma unroll
    for (int it = 0; it < 4; ++it) {
        int idx = tid + it * 256;        // float4 index 0..1023
        int r   = idx >> 6;              // row 0..15
        int c4  = idx & 63;              // float4 column
        float4 v = ((const float4*)out)[(long long)(row0 + r) * 64 + c4];
        float nd = norm_dst[row0 + r];
        float* t = &tile[r * LDS_PITCH + c4 * 4];
        t[0] = v.x * nd; t[1] = v.y * nd; t[2] = v.z * nd; t[3] = v.w * nd;
    }
    __syncthreads();

    const int lane  = tid & 31;
    const int wave  = tid >> 5;
    const int col0a = wave * 16;         // first column tile
    const int col0b = col0a + 128;       // second column tile
    const int n     = lane & 15;
    const int koff  = (lane >> 4) << 1;  // lanes 0-15: K+0,K+1 ; lanes 16-31: K+2,K+3

    v8f acc0 = {};
    v8f acc1 = {};
    const float* arow = &tile[(lane & 15) * LDS_PITCH + koff];
    const float* wa   = Wt + (long long)(col0a + n) * DIM + koff;
    const float* wb   = Wt + (long long)(col0b + n) * DIM + koff;

    for (int k = 0; k < DIM; k += 4) {
        v2f a  = *(const v2f*)(arow + k);   // A fragment (16x4 f32 layout)
        v2f b0 = *(const v2f*)(wa + k);     // B fragment tile a
        v2f b1 = *(const v2f*)(wb + k);     // B fragment tile b
        acc0 = __builtin_amdgcn_wmma_f32_16x16x4_f32(false, a, false, b0,
                                                     (short)0, acc0, false, false);
        acc1 = __builtin_amdgcn_wmma_f32_16x16x4_f32(false, a, false, b1,
                                                     (short)0, acc1, false, false);
    }

    // epilogue: bias + ELU, write back over the same rows (tile already staged in LDS)
    const int mbase = (lane >> 4) ? 8 : 0;
    float bva = bias[col0a + n];
    float bvb = bias[col0b + n];
    #pragma unroll
    for (int v = 0; v < 8; ++v) {
        int row = row0 + mbase + v;
        float x = acc0[v] + bva;
        out[(long long)row * DIM + col0a + n] = (x > 0.0f) ? x : (expf(x) - 1.0f);
        float y = acc1[v] + bvb;
        out[(long long)row * DIM + col0b + n] = (y > 0.0f) ? y : (expf(y) - 1.0f);
    }
}

extern "C" void kernel_launch(void* const* d_in, const int* in_sizes, int n_in,
                              void* d_out, int out_size, void* d_ws, size_t ws_size,
                              hipStream_t stream) {
    const float* h    = (const float*)d_in[0];
    const float* W    = (const float*)d_in[1];
    const float* bias = (const float*)d_in[2];
    const int*   src  = (const int*)d_in[3];
    const int*   dst  = (const int*)d_in[4];
    float* out = (float*)d_out;

    float* deg    = (float*)d_ws;            // [0,   N)  out_deg  -> norm_src
    float* outdeg = deg;                     // [N,  2N)  in_deg   -> norm_dst
    float* indeg  = deg + N_NODES;
    float* Wt     = deg + 2 * N_NODES;       // 256x256 transposed weight

    const int n_out = N_NODES * DIM;         // 25.6M — d_out doubles as agg accumulator

    zero_kernel<<<(n_out + 255) / 256, 256, 0, stream>>>(out, n_out, deg, 2 * N_NODES);
    degree_kernel<<<(N_EDGES + 255) / 256, 256, 0, stream>>>(src, dst, outdeg, indeg, N_EDGES);
    norm_kernel<<<(2 * N_NODES + 255) / 256, 256, 0, stream>>>(deg, 2 * N_NODES);
    transpose_kernel<<<(DIM * DIM) / 256, 256, 0, stream>>>(W, Wt);

    // 64 threads per edge (4 floats each) -> E*64 threads
    long long scatter_threads = (long long)N_EDGES * 64;
    scatter_kernel<<<(unsigned)(scatter_threads / 256), 256, 0, stream>>>(
        (const float4*)h, src, dst, outdeg, out, N_EDGES);

    gemm_elu_kernel<<<N_NODES / 16, 256, 0, stream>>>(out, Wt, bias, indeg);
}